// DualMamba_68719476736581
// MI455X (gfx1250) — compile-verified
//
#include <hip/hip_runtime.h>
#include <math.h>

// ---------------------------------------------------------------------------
// DualMamba for MI455X (gfx1250).  All GEMMs use V_WMMA_F32_16X16X4_F32
// (full-fp32 matrix op, wave32, one wave per 16x16 C tile).  Selective scan
// keeps the 64x64 state in registers, one thread per head-dim element.
// ---------------------------------------------------------------------------

typedef float v2f __attribute__((ext_vector_type(2)));
typedef float v8f __attribute__((ext_vector_type(8)));

#define B_SZ      2
#define D_MODEL   512
#define D_INNER   1024
#define D_STATE   64
#define NHEADS    16
#define HEADDIM   64
#define CONV_DIM  1152           // D_INNER + 2*D_STATE
#define D_PROJ    2192           // 2*D_INNER + 2*D_STATE + NHEADS
#define IMG       256

// ---------------------------------------------------------------------------
// Generic fp32 WMMA GEMM:  C[M,N] = A[M,K] @ W[N,K]^T (+ bias[n]) (+ pos[l,n])
// M,N multiples of 16; K multiple of 4.  One wave (32 lanes) per 16x16 tile.
// A,W rows assumed 8-byte aligned (all K here are even).
// Fragment layout (ISA 7.12.2): element j of lane l has k = 2*(l>>4)+j,
// A row = m0 + (l&15), W row(=B col) = n0 + (l&15).
// D layout: c[r] of lane l -> C[m0 + r + 8*(l>>4)][n0 + (l&15)].
// ---------------------------------------------------------------------------
__global__ __launch_bounds__(32) void gemm_f32_wmma(
    const float* __restrict__ A, const float* __restrict__ W,
    const float* __restrict__ bias, const float* __restrict__ pos,
    float* __restrict__ C, int M, int N, int K, int L)
{
  const int lane = threadIdx.x;
  const int r    = lane & 15;
  const int hi   = lane >> 4;
  const int n0   = blockIdx.x * 16;
  const int m0   = blockIdx.y * 16;

  const float* arow = A + (size_t)(m0 + r) * K + (hi << 1);
  const float* wrow = W + (size_t)(n0 + r) * K + (hi << 1);

  v8f c = {0.f, 0.f, 0.f, 0.f, 0.f, 0.f, 0.f, 0.f};

#pragma unroll 8
  for (int k = 0; k < K; k += 4) {
    v2f a = *(const v2f*)(arow + k);
    v2f b = *(const v2f*)(wrow + k);
    // D = A(16x4,f32) x B(4x16,f32) + C   -> v_wmma_f32_16x16x4_f32
    c = __builtin_amdgcn_wmma_f32_16x16x4_f32(
        /*neg_a=*/false, a, /*neg_b=*/false, b,
        /*c_mod=*/(short)0, c, /*reuse_a=*/false, /*reuse_b=*/false);
  }

  const int col = n0 + r;
  const float badd = bias ? bias[col] : 0.f;
#pragma unroll
  for (int i = 0; i < 8; ++i) {
    const int row = m0 + i + (hi << 3);
    float v = c[i] + badd;
    if (pos) v += pos[(size_t)(row % L) * N + col];
    C[(size_t)row * N + col] = v;
  }
}

// ---------------------------------------------------------------------------
// im2col for the strided patch-embed conv: X[(b*L+l), k], k = (c, ph, pw)
// ---------------------------------------------------------------------------
__global__ void im2col_kernel(const float* __restrict__ x, float* __restrict__ X,
                              int P, int L, int Kp, int Wg, int total)
{
  int idx = blockIdx.x * blockDim.x + threadIdx.x;
  if (idx >= total) return;
  int k  = idx % Kp;
  int bl = idx / Kp;
  int l  = bl % L;
  int b  = bl / L;
  int pp = P * P;
  int cch = k / pp;
  int rem = k % pp;
  int ph = rem / P, pw = rem % P;
  int gh = l / Wg, gw = l % Wg;
  X[idx] = x[(((size_t)b * 3 + cch) * IMG + gh * P + ph) * IMG + gw * P + pw];
}

// ---------------------------------------------------------------------------
// Depthwise causal conv (width 4) + bias + SiLU over the xBC slice of zxbcdt.
// ---------------------------------------------------------------------------
__global__ void conv_silu_kernel(const float* __restrict__ zx,
                                 const float* __restrict__ cw,
                                 const float* __restrict__ cb,
                                 float* __restrict__ xbc, int BL, int L)
{
  int idx = blockIdx.x * blockDim.x + threadIdx.x;
  if (idx >= BL * CONV_DIM) return;
  int c  = idx % CONV_DIM;
  int bl = idx / CONV_DIM;
  int l  = bl % L;
  int b  = bl / L;
  float acc = cb[c];
#pragma unroll
  for (int j = 0; j < 4; ++j) {
    int ls = l - 3 + j;
    if (ls >= 0)
      acc += cw[c * 4 + j] * zx[(size_t)(b * L + ls) * D_PROJ + D_INNER + c];
  }
  xbc[(size_t)bl * CONV_DIM + c] = acc / (1.f + expf(-acc));   // SiLU
}

// ---------------------------------------------------------------------------
// dt = softplus(dt_raw + dt_bias); dA = exp(dt * (-exp(A_log)))
// ---------------------------------------------------------------------------
__global__ void dt_kernel(const float* __restrict__ zx,
                          const float* __restrict__ dt_bias,
                          const float* __restrict__ A_log,
                          float* __restrict__ dt_sp, float* __restrict__ dA,
                          int BL)
{
  int idx = blockIdx.x * blockDim.x + threadIdx.x;
  if (idx >= BL * NHEADS) return;
  int h  = idx & (NHEADS - 1);
  int bl = idx >> 4;
  float v  = zx[(size_t)bl * D_PROJ + (D_PROJ - NHEADS) + h] + dt_bias[h];
  float sp = (v > 20.f) ? v : log1pf(expf(v));
  dt_sp[idx] = sp;
  dA[idx]    = expf(sp * (-expf(A_log[h])));
}

// ---------------------------------------------------------------------------
// Selective scan.  One block per (b, head); 64 threads; thread p owns
// h[p][n=0..63] in registers.  B/C/dt/dA staged CHUNK steps at a time in LDS.
//   h = dA*h + (dt*xh[p]) * B[n];  y[p] = sum_n h*C[n] + D*xh[p]
// ---------------------------------------------------------------------------
#define SCAN_CHUNK 16
__global__ __launch_bounds__(64) void scan_kernel(
    const float* __restrict__ xbc, const float* __restrict__ dt_sp,
    const float* __restrict__ dA, const float* __restrict__ Dp,
    float* __restrict__ y, int L)
{
  const int h = blockIdx.x % NHEADS;
  const int b = blockIdx.x / NHEADS;
  const int p = threadIdx.x;

  __shared__ float Bs[SCAN_CHUNK][D_STATE];
  __shared__ float Cs[SCAN_CHUNK][D_STATE];
  __shared__ float dts[SCAN_CHUNK];
  __shared__ float dAs[SCAN_CHUNK];

  float hstate[D_STATE];
#pragma unroll
  for (int n = 0; n < D_STATE; ++n) hstate[n] = 0.f;
  const float Dh = Dp[h];

  for (int l0 = 0; l0 < L; l0 += SCAN_CHUNK) {
    for (int s = 0; s < SCAN_CHUNK; ++s) {
      size_t base = (size_t)(b * L + l0 + s) * CONV_DIM;
      Bs[s][p] = xbc[base + D_INNER + p];
      Cs[s][p] = xbc[base + D_INNER + D_STATE + p];
    }
    if (p < SCAN_CHUNK) {
      size_t q = (size_t)(b * L + l0 + p) * NHEADS + h;
      dts[p] = dt_sp[q];
      dAs[p] = dA[q];
    }
    __syncthreads();

    for (int s = 0; s < SCAN_CHUNK; ++s) {
      const int l = l0 + s;
      const float xv  = xbc[(size_t)(b * L + l) * CONV_DIM + h * HEADDIM + p];
      const float dtx = dts[s] * xv;
      const float da  = dAs[s];
      float acc = 0.f;
#pragma unroll
      for (int n = 0; n < D_STATE; ++n) {
        hstate[n] = da * hstate[n] + dtx * Bs[s][n];
        acc += hstate[n] * Cs[s][n];
      }
      y[(size_t)(b * L + l) * D_INNER + h * HEADDIM + p] = acc + Dh * xv;
    }
    __syncthreads();
  }
}

// ---------------------------------------------------------------------------
// g = y * silu(z);  g *= rsqrt(mean(g^2)+1e-5) * norm_w   (in-place on y)
// One block per (b,l) row, 256 threads x 4 elements.
// ---------------------------------------------------------------------------
__global__ __launch_bounds__(256) void gate_norm_kernel(
    const float* __restrict__ zx, float* __restrict__ y,
    const float* __restrict__ nw)
{
  const int bl = blockIdx.x;
  const int t  = threadIdx.x;
  __shared__ float red[256];

  float g[4];
  float ss = 0.f;
#pragma unroll
  for (int i = 0; i < 4; ++i) {
    int d = t + i * 256;
    float z  = zx[(size_t)bl * D_PROJ + d];
    float gv = y[(size_t)bl * D_INNER + d] * (z / (1.f + expf(-z)));
    g[i] = gv;
    ss += gv * gv;
  }
  red[t] = ss;
  __syncthreads();
  for (int ofs = 128; ofs > 0; ofs >>= 1) {
    if (t < ofs) red[t] += red[t + ofs];
    __syncthreads();
  }
  const float sc = 1.0f / sqrtf(red[0] / (float)D_INNER + 1e-5f);
#pragma unroll
  for (int i = 0; i < 4; ++i) {
    int d = t + i * 256;
    y[(size_t)bl * D_INNER + d] = g[i] * sc * nw[d];
  }
}

// ---------------------------------------------------------------------------
// Un-patchify: img[b,c,gh*P+ph,gw*P+pw] = yp[b, l, (ph*P+pw)*3 + c]
// ---------------------------------------------------------------------------
__global__ void unpatch_kernel(const float* __restrict__ yp, float* __restrict__ img,
                               int P, int L, int Kp, int Wg, int total)
{
  int idx = blockIdx.x * blockDim.x + threadIdx.x;
  if (idx >= total) return;
  int w  = idx % IMG;
  int hh = (idx / IMG) % IMG;
  int c  = (idx / (IMG * IMG)) % 3;
  int b  = idx / (3 * IMG * IMG);
  int gh = hh / P, ph = hh % P, gw = w / P, pw = w % P;
  int l = gh * Wg + gw;
  int k = (ph * P + pw) * 3 + c;
  img[idx] = yp[(size_t)(b * L + l) * Kp + k];
}

// ---------------------------------------------------------------------------
// Final 1x1 conv over the 6-channel concat of the two branch images.
// ---------------------------------------------------------------------------
__global__ void combine_kernel(const float* __restrict__ x1, const float* __restrict__ x2,
                               const float* __restrict__ fw, const float* __restrict__ fb,
                               float* __restrict__ out, int total)
{
  int idx = blockIdx.x * blockDim.x + threadIdx.x;
  if (idx >= total) return;
  int hw = idx % (IMG * IMG);
  int o  = (idx / (IMG * IMG)) % 3;
  int b  = idx / (3 * IMG * IMG);
  float acc = fb[o];
#pragma unroll
  for (int c = 0; c < 3; ++c) {
    acc += fw[o * 6 + c]     * x1[((size_t)b * 3 + c) * IMG * IMG + hw];
    acc += fw[o * 6 + 3 + c] * x2[((size_t)b * 3 + c) * IMG * IMG + hw];
  }
  out[idx] = acc;
}

// ---------------------------------------------------------------------------
// Host-side driver
// ---------------------------------------------------------------------------
struct BlkP {
  const float *inW, *conv_w, *conv_b, *dt_bias, *A_log, *D, *norm_w, *outW;
};

static inline int ceil_div(int a, int b) { return (a + b - 1) / b; }

static void run_branch(const float* x, const float* w, const float* bias,
                       const float* pos, const BlkP* blk, const float* flw,
                       const float* flb, int P, float* ws, float* img_out,
                       hipStream_t stream)
{
  const int Wg = IMG / P;
  const int L  = Wg * Wg;
  const int BL = B_SZ * L;
  const int Kp = 3 * P * P;

  // workspace layout (floats); sized for the larger branch (L=1024)
  float* ws_t   = ws;                       // 2*1024*512
  float* ws_x   = ws_t   + 1048576;         // 2*1024*192 == 2*256*768
  float* ws_z   = ws_x   + 393216;          // 2*1024*2192
  float* ws_xbc = ws_z   + 4489216;         // 2*1024*1152
  float* ws_dt  = ws_xbc + 2359296;         // 2*1024*16
  float* ws_dA  = ws_dt  + 32768;           // 2*1024*16
  float* ws_y   = ws_dA  + 32768;           // 2*1024*1024

  const int TPB = 256;

  // 1. im2col + patch-embed GEMM (adds bias + positional encoding)
  int tot = BL * Kp;
  im2col_kernel<<<ceil_div(tot, TPB), TPB, 0, stream>>>(x, ws_x, P, L, Kp, Wg, tot);
  gemm_f32_wmma<<<dim3(D_MODEL / 16, BL / 16), 32, 0, stream>>>(
      ws_x, w, bias, pos, ws_t, BL, D_MODEL, Kp, L);

  // 2. Mamba2 blocks
  for (int i = 0; i < 2; ++i) {
    const BlkP& p = blk[i];
    gemm_f32_wmma<<<dim3(D_PROJ / 16, BL / 16), 32, 0, stream>>>(
        ws_t, p.inW, nullptr, nullptr, ws_z, BL, D_PROJ, D_MODEL, L);
    conv_silu_kernel<<<ceil_div(BL * CONV_DIM, TPB), TPB, 0, stream>>>(
        ws_z, p.conv_w, p.conv_b, ws_xbc, BL, L);
    dt_kernel<<<ceil_div(BL * NHEADS, TPB), TPB, 0, stream>>>(
        ws_z, p.dt_bias, p.A_log, ws_dt, ws_dA, BL);
    scan_kernel<<<B_SZ * NHEADS, 64, 0, stream>>>(
        ws_xbc, ws_dt, ws_dA, p.D, ws_y, L);
    gate_norm_kernel<<<BL, 256, 0, stream>>>(ws_z, ws_y, p.norm_w);
    gemm_f32_wmma<<<dim3(D_MODEL / 16, BL / 16), 32, 0, stream>>>(
        ws_y, p.outW, nullptr, nullptr, ws_t, BL, D_MODEL, D_INNER, L);
  }

  // 3. final linear + un-patchify (reuse im2col buffer for patch output)
  gemm_f32_wmma<<<dim3(Kp / 16, BL / 16), 32, 0, stream>>>(
      ws_t, flw, flb, nullptr, ws_x, BL, Kp, D_MODEL, L);
  tot = B_SZ * 3 * IMG * IMG;
  unpatch_kernel<<<ceil_div(tot, TPB), TPB, 0, stream>>>(ws_x, img_out, P, L, Kp, Wg, tot);
}

extern "C" void kernel_launch(void* const* d_in, const int* in_sizes, int n_in,
                              void* d_out, int out_size, void* d_ws, size_t ws_size,
                              hipStream_t stream)
{
  (void)in_sizes; (void)n_in; (void)out_size; (void)ws_size;
  const float* x    = (const float*)d_in[0];
  const float* w1   = (const float*)d_in[1];
  const float* b1   = (const float*)d_in[2];
  const float* w2   = (const float*)d_in[3];
  const float* b2   = (const float*)d_in[4];
  const float* pos1 = (const float*)d_in[5];
  const float* pos2 = (const float*)d_in[6];

  BlkP blocks1[2], blocks2[2];
  for (int i = 0; i < 2; ++i) {
    const float* const* q1 = (const float* const*)(d_in + 7 + i * 8);
    blocks1[i] = BlkP{q1[0], q1[1], q1[2], q1[3], q1[4], q1[5], q1[6], q1[7]};
    const float* const* q2 = (const float* const*)(d_in + 23 + i * 8);
    blocks2[i] = BlkP{q2[0], q2[1], q2[2], q2[3], q2[4], q2[5], q2[6], q2[7]};
  }
  const float* f1w = (const float*)d_in[39];
  const float* f1b = (const float*)d_in[40];
  const float* f2w = (const float*)d_in[41];
  const float* f2b = (const float*)d_in[42];
  const float* fw  = (const float*)d_in[43];
  const float* fb  = (const float*)d_in[44];

  float* ws = (float*)d_ws;
  float* img1 = ws + 11239424 - 786432;          // last two slots of layout
  float* img2 = img1 + 393216;

  run_branch(x, w1, b1, pos1, blocks1, f1w, f1b, 8,  ws, img1, stream);
  run_branch(x, w2, b2, pos2, blocks2, f2w, f2b, 16, ws, img2, stream);

  const int tot = B_SZ * 3 * IMG * IMG;
  combine_kernel<<<ceil_div(tot, 256), 256, 0, stream>>>(
      img1, img2, fw, fb, (float*)d_out, tot);
}